// DeepFMEmbedding_23441931501750
// MI455X (gfx1250) — compile-verified
//
#include <hip/hip_runtime.h>

// DeepFM embedding for MI455X (gfx1250, wave32).
//
// Shapes: N = B*L = 16384 samples, C = 39 channels (1 continuous + 38
// categorical), E = 16 embedding dim, VOCAB = 100000, two orders.
//
// Bandwidth analysis (the op is gather-bound, not compute-bound):
//   gathers from T1/T2: 2 * 16384*38*64B  = 79.7 MB  (random 64B rows)
//   output writes:      2 * 16384*39*64B  = 81.8 MB  (streaming, write-once)
//   Xi/Xv reads:        ~5 MB
//   => ~170 MB => ~7.3 us at 23.3 TB/s HBM.
//
// Kernel 1: embedding gather + scale (channels 1..38), 4 lanes per row,
//           b128 loads from T, non-temporal b128 stores.
// Kernel 2: the continuous channel is the only dense matrix op present
//           (rank-1 GEMM: out[s,e] = x_cont[s]*w[e] + b[e]); done with
//           V_WMMA_F32_16X16X4_F32, 16 samples per wave.

typedef __attribute__((ext_vector_type(2))) float v2f;
typedef __attribute__((ext_vector_type(4))) float v4f;
typedef __attribute__((ext_vector_type(8))) float v8f;

#define N_SAMPLES 16384   // B*L = 512*32
#define C_TOT     39
#define N_CAT     38
#define EDIM      16
#define VOCAB_SZ  100000

// ---------------------------------------------------------------------------
// Kernel 1: categorical gather, channels c = 1..38, both orders.
// tid -> (order from blockIdx.y, row = tid>>2, quad = tid&3).
// Each lane moves one v4f (16B); 4 consecutive lanes cover one 64B T row.
// Stores are fully coalesced over (s, c, e); marked non-temporal so the
// 82 MB write stream does not evict the T gather working set from L2.
// ---------------------------------------------------------------------------
__global__ __launch_bounds__(256) void deepfm_gather_kernel(
    const int*   __restrict__ Xi,
    const float* __restrict__ Xv,
    const float* __restrict__ T1,
    const float* __restrict__ T2,
    float*       __restrict__ out)
{
    const int order = blockIdx.y;
    const float* __restrict__ T = order ? T2 : T1;
    float* __restrict__ o = out + (size_t)order * (size_t)N_SAMPLES * C_TOT * EDIM;

    const int total = N_SAMPLES * N_CAT * 4;           // 2,490,368
    int tid = blockIdx.x * blockDim.x + threadIdx.x;
    if (tid >= total) return;

    const int q  = tid & 3;            // which float4 of the 16-float row
    const int r  = tid >> 2;           // row id in [0, N*38)
    const int s  = r / N_CAT;          // sample
    const int c1 = r % N_CAT;          // 0..37 -> channel c = c1+1

    const int   idx   = Xi[s * C_TOT + (c1 + 1)];      // D = 1
    const float scale = Xv[s * C_TOT + (c1 + 1)];

    // T layout: (F_CAT=38, VOCAB, E); one row = 16 floats = 64B, 64B aligned.
    const v4f* __restrict__ src =
        (const v4f*)(T + ((size_t)c1 * VOCAB_SZ + (size_t)idx) * EDIM) + q;
    v4f v = *src;                                       // global_load_b128
    v *= scale;

    v4f* dst = (v4f*)(o + ((size_t)s * C_TOT + (c1 + 1)) * EDIM) + q;
    __builtin_nontemporal_store(v, dst);                // b128 store, TH=NT
}

// ---------------------------------------------------------------------------
// Kernel 2: continuous channel (c = 0) as a 16x16x4 f32 WMMA.
//   D[m,n] = x_cont[base+m] * w[n] + b[n], then scaled by Xv[base+m, 0].
// One wave (32 lanes) per 16-sample tile. All loads are branch-free so EXEC
// stays all-1s at the WMMA (ISA 7.12 requirement).
//
// VGPR layouts (ISA 7.12.2, 32-bit operands, wave32):
//   A (16x4):  lanes 0-15 hold M=lane, {VGPR0=K0, VGPR1=K1};
//              lanes 16-31 hold {K2, K3}.  Only K=0 is nonzero here.
//   B (4x16):  lanes 0-15 hold N=lane, {VGPR0=K0, VGPR1=K1}; K=0 row = w.
//   C/D 16x16: VGPR v, lane L -> M = v + 8*(L>=16), N = L&15.
// ---------------------------------------------------------------------------
__global__ __launch_bounds__(32) void deepfm_cont_wmma_kernel(
    const int*   __restrict__ Xi,
    const float* __restrict__ Xv,
    const float* __restrict__ w1,
    const float* __restrict__ b1,
    const float* __restrict__ w2,
    const float* __restrict__ b2,
    float*       __restrict__ out)
{
    const int order = blockIdx.y;
    const float* __restrict__ w  = order ? w2 : w1;
    const float* __restrict__ bl = order ? b2 : b1;
    float* __restrict__ o = out + (size_t)order * (size_t)N_SAMPLES * C_TOT * EDIM;

    const int lane = threadIdx.x;        // 0..31
    const int half = lane >> 4;          // 0: lanes 0-15, 1: lanes 16-31
    const int l16  = lane & 15;
    const int base = blockIdx.x * 16;    // sample tile base

    // Branch-free operand setup: every lane issues the same loads.
    const float xc = (float)Xi[(base + l16) * C_TOT];  // x_cont for sample
    const float wv = w[l16];
    const float bv = bl[l16];

    v2f a;                               // A: only K=0 column nonzero
    a.x = half ? 0.0f : xc;
    a.y = 0.0f;
    v2f bmat;                            // B: only K=0 row (= w) nonzero
    bmat.x = half ? 0.0f : wv;
    bmat.y = 0.0f;
    v8f c = { bv, bv, bv, bv, bv, bv, bv, bv };   // C[m,n] = b[n]

    // D = A*B + C  ->  v_wmma_f32_16x16x4_f32
    c = __builtin_amdgcn_wmma_f32_16x16x4_f32(
        /*neg_a=*/false, a, /*neg_b=*/false, bmat,
        /*c_mod=*/(short)0, c, /*reuse_a=*/false, /*reuse_b=*/false);

    #pragma unroll
    for (int v = 0; v < 8; ++v) {
        const int m = v + half * 8;
        const int s = base + m;
        const float val = c[v] * Xv[s * C_TOT];        // * xv[:, 0]
        o[(size_t)s * C_TOT * EDIM + l16] = val;       // channel 0 row
    }
}

extern "C" void kernel_launch(void* const* d_in, const int* in_sizes, int n_in,
                              void* d_out, int out_size, void* d_ws, size_t ws_size,
                              hipStream_t stream) {
    // setup_inputs() order: Xi, Xv, w1_lin, b1_lin, w2_lin, b2_lin, T1, T2
    const int*   Xi = (const int*)  d_in[0];
    const float* Xv = (const float*)d_in[1];
    const float* w1 = (const float*)d_in[2];
    const float* b1 = (const float*)d_in[3];
    const float* w2 = (const float*)d_in[4];
    const float* b2 = (const float*)d_in[5];
    const float* T1 = (const float*)d_in[6];
    const float* T2 = (const float*)d_in[7];
    float* out = (float*)d_out;   // [fm_first | fm_second], each N*39*16 f32

    // Gather: 2 orders x (N*38 rows x 4 lanes/row)
    const int total = N_SAMPLES * N_CAT * 4;
    dim3 ggrid((total + 255) / 256, 2);
    deepfm_gather_kernel<<<ggrid, dim3(256), 0, stream>>>(Xi, Xv, T1, T2, out);

    // Continuous channel: 2 orders x 1024 tiles of 16 samples, 1 wave each
    dim3 cgrid(N_SAMPLES / 16, 2);
    deepfm_cont_wmma_kernel<<<cgrid, dim3(32), 0, stream>>>(
        Xi, Xv, w1, b1, w2, b2, out);
}